// VisualBranch_vsgnet_52432960749729
// MI455X (gfx1250) — compile-verified
//
#include <hip/hip_runtime.h>
#include <hip/hip_bf16.h>
#include <math.h>

// Problem dims (from reference)
#define NN     128
#define BB     16
#define HW     2048
#define DOBJ   2048
#define DQ     512
#define DCTX   1024

typedef __attribute__((ext_vector_type(2))) float v2f;
typedef __attribute__((ext_vector_type(8))) float v8f;

// ---------------------------------------------------------------------------
// Kernel 0: group objects by batch. perm[] = object ids sorted by batch,
// base[b] = start offset of batch b in perm (base[BB] = NN). N=128: serial.
// ---------------------------------------------------------------------------
__global__ void group_kernel(const int* __restrict__ bidx,
                             int* __restrict__ perm,
                             int* __restrict__ base)
{
    if (threadIdx.x == 0 && blockIdx.x == 0) {
        int cnt[BB];
        for (int b = 0; b < BB; ++b) cnt[b] = 0;
        for (int n = 0; n < NN; ++n) cnt[bidx[n]]++;
        int acc = 0;
        for (int b = 0; b < BB; ++b) { base[b] = acc; acc += cnt[b]; }
        base[BB] = acc;
        int cur[BB];
        for (int b = 0; b < BB; ++b) cur[b] = base[b];
        for (int n = 0; n < NN; ++n) perm[cur[bidx[n]]++] = n;
    }
}

// ---------------------------------------------------------------------------
// Kernel 1: qg[m] = relu(obj[perm[m]] @ W1 + b1) using fp32 WMMA 16x16x4.
// One wave per 16x16 tile; gathers A rows through perm (fused row-permute).
// A-frag (16x4 f32): lanes 0-15 -> {K=k0,k0+1}; lanes 16-31 -> {K=k0+2,k0+3}
// B-frag (4x16 f32): same split on the K rows, col = lane&15
// C/D (16x16 f32): vgpr i -> M = i + 8*half, N = lane&15
// ---------------------------------------------------------------------------
__global__ __launch_bounds__(32)
void gemm1_relu_wmma(const float* __restrict__ A,     // [NN, DOBJ]
                     const float* __restrict__ W,     // [DOBJ, DQ]
                     const float* __restrict__ bias,  // [DQ]
                     const int*   __restrict__ perm,  // [NN]
                     float* __restrict__ Qg)          // [NN, DQ] (permuted rows)
{
    const int ntile = blockIdx.x;           // 0..31
    const int mtile = blockIdx.y;           // 0..7
    const int lane  = threadIdx.x;          // 0..31
    const int r     = lane & 15;
    const int half  = lane >> 4;

    const int src_row = perm[mtile * 16 + r];
    const float* __restrict__ Arow = A + (size_t)src_row * DOBJ;
    const int n0 = ntile * 16;

    v8f c = {0.f, 0.f, 0.f, 0.f, 0.f, 0.f, 0.f, 0.f};

    #pragma unroll 4
    for (int k0 = 0; k0 < DOBJ; k0 += 4) {
        const int ka = k0 + 2 * half;
        v2f a, b;
        a.x = Arow[ka];
        a.y = Arow[ka + 1];
        b.x = W[(size_t)ka * DQ + n0 + r];
        b.y = W[(size_t)(ka + 1) * DQ + n0 + r];
        c = __builtin_amdgcn_wmma_f32_16x16x4_f32(
                false, a, false, b, (short)0, c, false, false);
    }

    const float bv = bias[n0 + r];
    #pragma unroll
    for (int i = 0; i < 8; ++i) {
        const int m = mtile * 16 + i + 8 * half;
        float v = c[i] + bv;
        Qg[(size_t)m * DQ + n0 + r] = v > 0.f ? v : 0.f;
    }
}

// ---------------------------------------------------------------------------
// Kernel 2: batched attention logits via WMMA.
//   dots[m, h] = qg[m] . K[b, h, :]  for objects m in batch b's group.
// Grid = (batch, HW/16, max m-tiles). One wave per 16x16 tile, d-loop of 512.
// K[b] is streamed once per batch (~64 MB total vs 512 MB per-object).
// ---------------------------------------------------------------------------
__global__ __launch_bounds__(32)
void dot_wmma(const float* __restrict__ Qg,    // [NN, DQ] (permuted)
              const float* __restrict__ CK,    // [BB, HW, DQ]
              const int*   __restrict__ base,  // [BB+1]
              float* __restrict__ dots)        // [NN, HW] (permuted rows)
{
    const int b     = blockIdx.x;   // 0..15
    const int htile = blockIdx.y;   // 0..127
    const int mtile = blockIdx.z;   // 0..7

    const int lo = base[b];
    const int hi = base[b + 1];
    const int m0 = lo + mtile * 16;
    if (m0 >= hi) return;           // empty tile for this batch

    const int lane = threadIdx.x;
    const int r    = lane & 15;
    const int half = lane >> 4;

    // A rows: clamp padded rows to a valid address (results masked on store)
    const int arow = min(m0 + r, NN - 1);
    const float* __restrict__ Aq = Qg + (size_t)arow * DQ;
    // B cols: B[d][n] = K[b, htile*16+n, d]; this lane owns col n = r
    const float* __restrict__ Kc = CK + ((size_t)b * HW + htile * 16 + r) * DQ;

    v8f c = {0.f, 0.f, 0.f, 0.f, 0.f, 0.f, 0.f, 0.f};

    #pragma unroll 4
    for (int k0 = 0; k0 < DQ; k0 += 4) {
        const int ka = k0 + 2 * half;
        v2f a, bb;
        a.x  = Aq[ka];
        a.y  = Aq[ka + 1];
        bb.x = Kc[ka];
        bb.y = Kc[ka + 1];
        c = __builtin_amdgcn_wmma_f32_16x16x4_f32(
                false, a, false, bb, (short)0, c, false, false);
    }

    const int h = htile * 16 + r;
    #pragma unroll
    for (int i = 0; i < 8; ++i) {
        const int m = m0 + i + 8 * half;
        if (m < hi) dots[(size_t)m * HW + h] = c[i];
    }
}

// ---------------------------------------------------------------------------
// Kernel 3: softmax over HW + weighted sum with vals -> scal[perm[i]].
// One block (256 thr = 8 wave32) per permuted object; row staged in LDS.
// ---------------------------------------------------------------------------
__global__ __launch_bounds__(256)
void softmax_kernel(const float* __restrict__ dots,  // [NN, HW] (permuted)
                    const int*   __restrict__ perm,  // [NN]
                    const int*   __restrict__ bidx,  // [NN]
                    const float* __restrict__ CV,    // [BB, HW]
                    float* __restrict__ scal)        // [NN]
{
    __shared__ float row[HW];
    __shared__ float redA[8];
    __shared__ float redB[8];

    const int i    = blockIdx.x;     // permuted index
    const int tid  = threadIdx.x;
    const int wave = tid >> 5;
    const int lane = tid & 31;
    const int n    = perm[i];
    const int b    = bidx[n];
    const float* __restrict__ V = CV + (size_t)b * HW;

    for (int h = tid; h < HW; h += 256) row[h] = dots[(size_t)i * HW + h];
    __syncthreads();

    float lmax = -INFINITY;
    for (int h = tid; h < HW; h += 256) lmax = fmaxf(lmax, row[h]);
    #pragma unroll
    for (int off = 16; off > 0; off >>= 1)
        lmax = fmaxf(lmax, __shfl_xor(lmax, off, 32));
    if (lane == 0) redA[wave] = lmax;
    __syncthreads();
    float gmax = redA[0];
    #pragma unroll
    for (int w = 1; w < 8; ++w) gmax = fmaxf(gmax, redA[w]);
    __syncthreads();

    float se = 0.f, sv = 0.f;
    for (int h = tid; h < HW; h += 256) {
        const float e = __expf(row[h] - gmax);
        se += e;
        sv += e * V[h];
    }
    #pragma unroll
    for (int off = 16; off > 0; off >>= 1) {
        se += __shfl_xor(se, off, 32);
        sv += __shfl_xor(sv, off, 32);
    }
    if (lane == 0) { redA[wave] = se; redB[wave] = sv; }
    __syncthreads();
    if (tid == 0) {
        float tse = 0.f, tsv = 0.f;
        #pragma unroll
        for (int w = 0; w < 8; ++w) { tse += redA[w]; tsv += redB[w]; }
        scal[n] = tsv / tse;
    }
}

// ---------------------------------------------------------------------------
// Kernel 4: colsum[j] = sum_d W2[d, j]. (ctx rows are a broadcast scalar, so
// ctx@W2 == scal[n]*colsum[j] exactly -- rank-1 collapse of GEMM2.)
// ---------------------------------------------------------------------------
__global__ __launch_bounds__(256)
void colsum_kernel(const float* __restrict__ W2,   // [DQ, DCTX]
                   float* __restrict__ colsum)     // [DCTX]
{
    const int j = blockIdx.x * 256 + threadIdx.x;
    if (j < DCTX) {
        float s = 0.f;
        for (int d = 0; d < DQ; ++d) s += W2[(size_t)d * DCTX + j];
        colsum[j] = s;
    }
}

// ---------------------------------------------------------------------------
// Kernel 5: out[n, 0:2048] = obj[n, :]
//           out[n, 2048:3072] = relu(scal[n]*colsum[j] + b2[j])
// ---------------------------------------------------------------------------
__global__ __launch_bounds__(256)
void out_kernel(const float* __restrict__ obj,     // [NN, DOBJ]
                const float* __restrict__ scal,    // [NN]
                const float* __restrict__ colsum,  // [DCTX]
                const float* __restrict__ b2,      // [DCTX]
                float* __restrict__ out)           // [NN, DOBJ+DCTX]
{
    const int n   = blockIdx.x;
    const int tid = threadIdx.x;
    const float s = scal[n];

    const float4* __restrict__ src = (const float4*)(obj + (size_t)n * DOBJ);
    float4* __restrict__ dst = (float4*)(out + (size_t)n * (DOBJ + DCTX));
    for (int i = tid; i < DOBJ / 4; i += 256) dst[i] = src[i];

    float* __restrict__ tail = out + (size_t)n * (DOBJ + DCTX) + DOBJ;
    for (int j = tid; j < DCTX; j += 256) {
        const float v = s * colsum[j] + b2[j];
        tail[j] = v > 0.f ? v : 0.f;
    }
}

// ---------------------------------------------------------------------------
extern "C" void kernel_launch(void* const* d_in, const int* in_sizes, int n_in,
                              void* d_out, int out_size, void* d_ws, size_t ws_size,
                              hipStream_t stream) {
    const float* obj  = (const float*)d_in[0];   // [128, 2048]
    const int*   bidx = (const int*)  d_in[1];   // [128]
    const float* ck   = (const float*)d_in[2];   // [16, 2048, 512]
    const float* cv   = (const float*)d_in[3];   // [16, 2048]
    const float* W1   = (const float*)d_in[4];   // [2048, 512]
    const float* b1   = (const float*)d_in[5];   // [512]
    const float* W2   = (const float*)d_in[6];   // [512, 1024]
    const float* b2   = (const float*)d_in[7];   // [1024]
    float* out = (float*)d_out;                  // [128, 3072]

    char* ws = (char*)d_ws;
    float* qg     = (float*)(ws);                  // 128*512*4   = 256 KB
    float* dots   = (float*)(ws + 262144);         // 128*2048*4  = 1 MB
    float* scal   = (float*)(ws + 1310720);        // 512 B
    int*   perm   = (int*)  (ws + 1311232);        // 512 B
    int*   base   = (int*)  (ws + 1311744);        // 68 B
    float* colsum = (float*)(ws + 1312256);        // 4 KB

    // 0) group objects by batch (serial, N=128)
    group_kernel<<<1, 32, 0, stream>>>(bidx, perm, base);

    // 1) qg = relu(obj[perm] @ W1 + b1) -- fp32 WMMA, fused row-gather
    gemm1_relu_wmma<<<dim3(DQ / 16, NN / 16), 32, 0, stream>>>(obj, W1, b1, perm, qg);

    // 2) batched attention logits: WMMA tiles, K[b] streamed once per batch
    dot_wmma<<<dim3(BB, HW / 16, NN / 16), 32, 0, stream>>>(qg, ck, base, dots);

    // 3) softmax + weighted value sum -> scal
    softmax_kernel<<<NN, 256, 0, stream>>>(dots, perm, bidx, cv, scal);

    // 4) column sums of W2 (rank-1 collapse of GEMM2)
    colsum_kernel<<<(DCTX + 255) / 256, 256, 0, stream>>>(W2, colsum);

    // 5) assemble output
    out_kernel<<<NN, 256, 0, stream>>>(obj, scal, colsum, b2, out);
}